// SAETI_67903432950243
// MI455X (gfx1250) — compile-verified
//
#include <hip/hip_runtime.h>

typedef float v2f __attribute__((ext_vector_type(2)));
typedef float v4f __attribute__((ext_vector_type(4)));
typedef float v8f __attribute__((ext_vector_type(8)));

#define B_TOT 4096
#define S_TOT 256
#define F_TOT 64
#define K_TOT 16
#define L_TOT 256

#define BT    16           // b rows per workgroup (WMMA M)
#define FC    32           // f values per workgroup
#define SC    8            // s chunk staged in LDS (double-buffered)
#define NCHUNK (S_TOT / SC)
#define FPAD  36           // padded f-extent per (s,b) cell (16B-aligned, bank-spread)
#define NWAVE 8
#define FPW   (FC / NWAVE) // 4 f per wave
#define CPW   ((BT * SC * FC / 4) / 256)  // async b128 copies per wave-lane = 4

__global__ __launch_bounds__(256)
void saeti_cls_gather(const float* __restrict__ x,
                      const float* __restrict__ Wc,
                      const float* __restrict__ snip,
                      float* __restrict__ out)
{
    __shared__ __align__(16) float xs[2][SC][BT][FPAD]; // double-buffered x tile
    __shared__ float cs[NWAVE][16][16];                 // C spill for argmax
    __shared__ int   idxs[BT][FC];                      // argmax results

    const int tid  = threadIdx.x;
    const int wave = tid >> 5;
    const int lane = tid & 31;
    const int hb   = lane >> 4;   // half-wave: 0 -> lanes 0-15, 1 -> 16-31
    const int bl   = lane & 15;

    const int b0 = blockIdx.x * BT;
    const int f0 = blockIdx.y * FC;

    v8f acc[FPW];
    #pragma unroll
    for (int j = 0; j < FPW; ++j)
        #pragma unroll
        for (int r = 0; r < 8; ++r) acc[j][r] = 0.0f;

    // Per-lane W fragment base pointers (loads below become imm-offset forms)
    const float* wl[FPW];
    #pragma unroll
    for (int j = 0; j < FPW; ++j)
        wl[j] = Wc + ((size_t)(f0 + wave * FPW + j) * S_TOT + 2 * hb) * K_TOT + bl;

    // Async global->LDS staging of one (SC x BT x FC) x tile, b128 per lane.
    // buf is always a compile-time constant at the call sites -> LDS dest
    // offsets for both buffers are loop-invariant and get hoisted.
    auto issue_chunk = [&](int sc, int buf) {
        const float* base = x + ((size_t)b0 * S_TOT + (size_t)sc * SC) * F_TOT + f0;
        #pragma unroll
        for (int it = 0; it < CPW; ++it) {           // 4 async copies / lane
            const int i  = it * 256 + tid;
            const int f4 = i & 7;
            const int s  = (i >> 3) & (SC - 1);
            const int b  = i >> 6;
            unsigned lds_off = (unsigned)(uintptr_t)&xs[buf][s][b][f4 * 4];
            unsigned goff    = (unsigned)(((b * S_TOT + s) * F_TOT + f4 * 4) * 4);
            asm volatile("global_load_async_to_lds_b128 %0, %1, %2"
                         :: "v"(lds_off), "v"(goff), "s"(base) : "memory");
        }
    };

    // WMMA accumulate chunk n from buffer buf:
    // C[16b x 16k] += A[16b x 4s] * B[4s x 16k], A rows K=0,1 / K=2,3 per half-wave
    auto compute_chunk = [&](int n, int buf) {
        const int s0 = n * SC;
        #pragma unroll
        for (int j = 0; j < FPW; ++j) {
            const int fl = wave * FPW + j;
            #pragma unroll
            for (int ss = 0; ss < SC / 4; ++ss) {    // 2 WMMA steps
                const int sl = ss * 4 + 2 * hb;
                v2f a, bfr;
                float ax = xs[buf][sl    ][bl][fl];
                float ay = xs[buf][sl + 1][bl][fl];
                a.x = (ax == ax) ? ax : 0.0f;        // NaN -> 0 (reference semantics)
                a.y = (ay == ay) ? ay : 0.0f;
                bfr.x = wl[j][(s0 + ss * 4) * K_TOT];          // B[K=sl  , N=bl]
                bfr.y = wl[j][(s0 + ss * 4) * K_TOT + K_TOT];  // B[K=sl+1, N=bl]
                acc[j] = __builtin_amdgcn_wmma_f32_16x16x4_f32(
                    false, a, false, bfr, (short)0, acc[j], false, false);
            }
        }
    };

    issue_chunk(0, 0);  // prologue

    for (int n = 0; n < NCHUNK; n += 2) {
        // ---- phase A: prefetch n+1 into buf1, compute n from buf0 ----
        if (n > 0) __syncthreads();                  // buf1 free to overwrite
        issue_chunk(n + 1, 1);                       // n+1 <= NCHUNK-1 always
        // own chunk-n copies complete: allow only the 4 just-issued outstanding
        asm volatile("s_wait_asynccnt 0x4" ::: "memory");
        __syncthreads();                             // all waves' chunk-n writes visible
        compute_chunk(n, 0);

        // ---- phase B: prefetch n+2 into buf0, compute n+1 from buf1 ----
        __syncthreads();                             // buf0 free to overwrite
        if (n + 2 < NCHUNK) {
            issue_chunk(n + 2, 0);
            asm volatile("s_wait_asynccnt 0x4" ::: "memory");
        } else {
            asm volatile("s_wait_asynccnt 0x0" ::: "memory");  // final drain
        }
        __syncthreads();
        compute_chunk(n + 1, 1);
    }
    __syncthreads();

    // ---- argmax over k per (b,f): spill C to LDS, lanes 0-15 scan rows ----
    #pragma unroll
    for (int j = 0; j < FPW; ++j) {
        #pragma unroll
        for (int r = 0; r < 8; ++r)
            cs[wave][hb * 8 + r][bl] = acc[j][r];   // C: M = r + 8*hb, N = bl
        __syncthreads();
        if (lane < 16) {
            float best = cs[wave][lane][0];
            int   kb   = 0;
            #pragma unroll
            for (int k = 1; k < K_TOT; ++k) {
                float v = cs[wave][lane][k];
                if (v > best) { best = v; kb = k; }   // first-max tie-break
            }
            idxs[lane][wave * FPW + j] = kb;
        }
        __syncthreads();
    }

    // ---- gather + stream out: b128 loads, non-temporal b128 stores ----
    const v4f* snv = (const v4f*)snip;
    v4f*       ouv = (v4f*)out;
    const int  sub = tid >> 6;      // 4 rows in flight per iteration
    const int  tl  = tid & 63;      // float4 index within a 256-float row
    #pragma unroll 4
    for (int it = 0; it < (BT * FC) / 4; ++it) {    // 128 iters
        const int row = it * 4 + sub;
        const int bls = row >> 5;          // / FC
        const int fl  = row & (FC - 1);
        const int k   = idxs[bls][fl];
        v4f val = snv[((size_t)(f0 + fl) * K_TOT + k) * (L_TOT / 4) + tl];
        __builtin_nontemporal_store(
            val,
            ouv + (((size_t)(b0 + bls) * F_TOT) + (f0 + fl)) * (L_TOT / 4) + tl);
    }
}

extern "C" void kernel_launch(void* const* d_in, const int* in_sizes, int n_in,
                              void* d_out, int out_size, void* d_ws, size_t ws_size,
                              hipStream_t stream) {
    (void)in_sizes; (void)n_in; (void)d_ws; (void)ws_size; (void)out_size;
    const float* x    = (const float*)d_in[0];
    const float* Wc   = (const float*)d_in[1];
    const float* snip = (const float*)d_in[2];
    float* out = (float*)d_out;

    dim3 grid(B_TOT / BT, F_TOT / FC);   // (256, 2)
    saeti_cls_gather<<<grid, 256, 0, stream>>>(x, Wc, snip, out);
}